// HierarchicalPattern_55490977465135
// MI455X (gfx1250) — compile-verified
//
#include <hip/hip_runtime.h>
#include <hip/hip_bf16.h>
#include <math.h>

// ---------------------------------------------------------------------------
// HierarchicalPattern sparse-attention mask builder for MI455X (gfx1250).
//   lf = x @ W_local^T ; gf = x @ W_global^T    (f32 WMMA 16x16x4)
//   local top-ks of relu(lf[q].lf[win]) and global top-32 of relu(gf gf^T)
//   out (B,1,S,S): -inf everywhere, 0.0 at selected positions.
// f32 WMMA is used (not bf16/fp8): output is discrete 0/-inf so score
// ranking must match the f32 reference; roofline is dominated by the
// 268 MB output stream (~12us at 23.3 TB/s), compute is ~21 GFLOP.
// The -inf fill uses non-temporal b128 stores so the 268 MB stream does
// not evict the L2-resident gf working set (1 MB/batch, re-read ~256x).
// ---------------------------------------------------------------------------

typedef __attribute__((ext_vector_type(2))) float v2f;
typedef __attribute__((ext_vector_type(4))) float v4f;
typedef __attribute__((ext_vector_type(8))) float v8f;

#define S_LEN  4096
#define BATCH  4
#define DMODEL 1024
#define FDIM   64
#define LWIN   16
#define GK     32
#define CHUNK  256

__device__ __forceinline__ v8f wmma_f32_k4(v2f a, v2f b, v8f c) {
  // D = A(16x4 f32) * B(4x16 f32) + C(16x16 f32)
  return __builtin_amdgcn_wmma_f32_16x16x4_f32(false, a, false, b,
                                               (short)0, c, false, false);
}

// ---------------------------------------------------------------------------
// Kernel 1: fill output with -inf, non-temporal b128 streaming stores.
// ---------------------------------------------------------------------------
__global__ void fill_neginf_kernel(v4f* __restrict__ out, long n4) {
  const float NI = -__builtin_huge_valf();
  v4f v = {NI, NI, NI, NI};
  long i = (long)blockIdx.x * blockDim.x + threadIdx.x;
  long stride = (long)gridDim.x * blockDim.x;
  for (; i < n4; i += stride)
    __builtin_nontemporal_store(v, &out[i]);   // global_store_b128 ... nt
}

// ---------------------------------------------------------------------------
// Kernel 2: projections lf/gf = x @ W^T via V_WMMA_F32_16X16X4_F32.
// Block = 128 threads (4 waves); block handles 16 rows x all 64 features;
// wave w handles feature cols [16w, 16w+16). K loop: 256 steps of K=4.
// ---------------------------------------------------------------------------
__global__ __launch_bounds__(128) void proj_kernel(
    const float* __restrict__ x, const float* __restrict__ Wl,
    const float* __restrict__ Wg, float* __restrict__ lf,
    float* __restrict__ gf) {
  const int r0   = blockIdx.x * 16;        // output row tile (flattened B*S)
  const int wave = threadIdx.x >> 5;
  const int lane = threadIdx.x & 31;
  const int half = lane >> 4;              // 0: K0/1, 1: K2/3
  const int lid  = lane & 15;
  const int colbase = wave * 16;

  const float* xrow  = x  + (size_t)(r0 + lid)      * DMODEL + half * 2;
  const float* wlrow = Wl + (size_t)(colbase + lid) * DMODEL + half * 2;
  const float* wgrow = Wg + (size_t)(colbase + lid) * DMODEL + half * 2;

  v8f accl = {}; v8f accg = {};
  for (int k0 = 0; k0 < DMODEL; k0 += 4) {
    v2f a  = *(const v2f*)(xrow  + k0);    // A[m=lid][k0+2*half+{0,1}]
    v2f bl = *(const v2f*)(wlrow + k0);    // B[k][n=lid] = W[n][k]
    v2f bg = *(const v2f*)(wgrow + k0);
    accl = wmma_f32_k4(a, bl, accl);
    accg = wmma_f32_k4(a, bg, accg);
  }
  // D layout: VGPR v -> M = v + 8*half, N = lid
  for (int v = 0; v < 8; ++v) {
    size_t o = (size_t)(r0 + v + 8 * half) * FDIM + colbase + lid;
    lf[o] = accl[v];
    gf[o] = accg[v];
  }
}

// ---------------------------------------------------------------------------
// Sorted top-32 list distributed one element per lane (lane 0 = best).
// Order: value desc, index asc  (replicates jax.lax.top_k stable tie-break).
// ---------------------------------------------------------------------------
__device__ __forceinline__ void list_insert(float& lv, int& li, float cv,
                                            int cj, int lane) {
  bool beat = (lv > cv) || (lv == cv && li < cj);  // existing ranks above cand
  float upv = __shfl_up(lv, 1);
  int   upi = __shfl_up(li, 1);
  int   pb  = __shfl_up((int)beat, 1);
  bool at_ins = !beat && (lane == 0 || pb != 0);   // insertion slot
  lv = beat ? lv : (at_ins ? cv : upv);
  li = beat ? li : (at_ins ? cj : upi);
}

__device__ __forceinline__ void topk_scan(const float* __restrict__ srow,
                                          int jbase, float& lv, int& li,
                                          int lane) {
  for (int r = 0; r < CHUNK / 32; ++r) {
    float cv = srow[r * 32 + lane];
    int   cj = jbase + r * 32 + lane;
    float wv = __shfl(lv, 31);
    int   wi = __shfl(li, 31);
    // banned entries are stored as -1.0f; real scores are >= 0 (post-relu)
    bool qual = (cv >= 0.0f) && ((cv > wv) || (cv == wv && cj < wi));
    unsigned m = (unsigned)__ballot(qual);
    while (m) {                       // process candidates in increasing j
      int src = __ffs(m) - 1;
      m &= m - 1;
      float ccv = __shfl(cv, src);    // uniform across wave
      int   ccj = __shfl(cj, src);
      float wv2 = __shfl(lv, 31);
      int   wi2 = __shfl(li, 31);
      if ((ccv > wv2) || (ccv == wv2 && ccj < wi2))
        list_insert(lv, li, ccv, ccj, lane);
    }
  }
}

// ---------------------------------------------------------------------------
// Local branch: top-ks of relu(lf[q] . lf[q-15..q]) with lowest-w tie-break.
// ---------------------------------------------------------------------------
__device__ __forceinline__ void local_select(float* __restrict__ outrow,
                                             int qg, int q0,
                                             const float* __restrict__ lfb,
                                             const float Lq[16][FDIM],
                                             int lane) {
  float s = -1.0f;
  int p = qg - (LWIN - 1) + lane;
  if (lane < LWIN && p >= 0) {
    const float* fq = &Lq[qg - q0][0];
    const float* fp = (p >= q0 && p < q0 + 16) ? &Lq[p - q0][0]
                                               : (lfb + (size_t)p * FDIM);
    float acc = 0.0f;
    for (int k = 0; k < FDIM; ++k) acc += fq[k] * fp[k];
    s = fmaxf(acc, 0.0f);
  }
  int L  = (qg + 1 < LWIN) ? qg + 1 : LWIN;
  int ks = (int)((double)L * (1.0 - 0.8));  // Python: int(L*0.1999999999999...)
  if (ks < 1) ks = 1;
  for (int it = 0; it < ks; ++it) {
    float bv = s; int bw = lane;
    for (int off = 16; off > 0; off >>= 1) {  // argmax, lower-w tie-break
      float ov = __shfl_xor(bv, off);
      int   ow = __shfl_xor(bw, off);
      if (ov > bv || (ov == bv && ow < bw)) { bv = ov; bw = ow; }
    }
    if (lane == 0) outrow[qg - (LWIN - 1) + bw] = 0.0f;
    if (lane == bw) s = -2.0f;               // exclude winner
  }
}

// ---------------------------------------------------------------------------
// Kernel 3: per 16-query tile: scores = relu(Gq . gf^T) via WMMA into LDS
// chunks of 16x256, streaming stable top-32 per query, then scatter zeros.
// Block = 256 threads (8 waves); wave w owns queries 2w, 2w+1 of the tile.
// ---------------------------------------------------------------------------
__global__ __launch_bounds__(256) void hier_mask_kernel(
    const float* __restrict__ lf, const float* __restrict__ gf,
    float* __restrict__ out) {
  __shared__ float Gq[16][FDIM];         // 4 KB
  __shared__ float Lq[16][FDIM];         // 4 KB
  __shared__ float Sc[16][CHUNK];        // 16 KB score chunk

  const int b    = blockIdx.y;
  const int q0   = blockIdx.x * 16;
  const int tid  = threadIdx.x;
  const int wave = tid >> 5;
  const int lane = tid & 31;
  const int half = lane >> 4;
  const int lid  = lane & 15;

  const float* lfb = lf + (size_t)b * S_LEN * FDIM;
  const float* gfb = gf + (size_t)b * S_LEN * FDIM;

  for (int i = tid; i < 16 * FDIM; i += 256) {
    int r = i >> 6, k = i & 63;
    Gq[r][k] = gfb[(size_t)(q0 + r) * FDIM + k];
    Lq[r][k] = lfb[(size_t)(q0 + r) * FDIM + k];
  }
  __syncthreads();

  float lv0 = -1.0f, lv1 = -1.0f;        // sentinel: below any relu score
  int   li0 = 0x7fffffff, li1 = 0x7fffffff;

  for (int jbase = 0; jbase < S_LEN; jbase += CHUNK) {
    // ---- score chunk via WMMA: wave w computes j-tiles {w, w+8} -----------
    for (int t = wave; t < CHUNK / 16; t += 8) {
      int jt = jbase + t * 16;
      v8f acc = {};
      const float* grow = gfb + (size_t)(jt + lid) * FDIM + half * 2;
      for (int k0 = 0; k0 < FDIM; k0 += 4) {
        v2f a  = *(const v2f*)&Gq[lid][k0 + half * 2];  // A[m=lid][k]
        v2f bb = *(const v2f*)(grow + k0);              // B[k][n=lid]=gf[j][k]
        acc = wmma_f32_k4(a, bb, acc);
      }
      int jg = jt + lid;
      for (int v = 0; v < 8; ++v) {
        int qi = v + 8 * half;
        int qg = q0 + qi;
        bool banned = (jg <= qg) && (jg >= qg - (LWIN - 1));
        Sc[qi][t * 16 + lid] = banned ? -1.0f : fmaxf(acc[v], 0.0f);
      }
    }
    __syncthreads();
    // ---- streaming stable top-32 for this wave's two queries --------------
    topk_scan(&Sc[2 * wave][0],     jbase, lv0, li0, lane);
    topk_scan(&Sc[2 * wave + 1][0], jbase, lv1, li1, lane);
    __syncthreads();
  }

  // ---- scatter: 32 global zeros + local top-ks zeros per owned query ------
  {
    int qg = q0 + 2 * wave;
    float* row = out + (size_t)b * S_LEN * S_LEN + (size_t)qg * S_LEN;
    row[li0] = 0.0f;                                   // one index per lane
    local_select(row, qg, q0, lfb, Lq, lane);

    qg = q0 + 2 * wave + 1;
    row = out + (size_t)b * S_LEN * S_LEN + (size_t)qg * S_LEN;
    row[li1] = 0.0f;
    local_select(row, qg, q0, lfb, Lq, lane);
  }
}

// ---------------------------------------------------------------------------
extern "C" void kernel_launch(void* const* d_in, const int* in_sizes, int n_in,
                              void* d_out, int out_size, void* d_ws,
                              size_t ws_size, hipStream_t stream) {
  const float* x  = (const float*)d_in[0];
  const float* Wl = (const float*)d_in[1];
  // d_in[2] = W_medium: unused by the reference output (mf is dead code)
  const float* Wg = (const float*)d_in[3];
  float* out = (float*)d_out;

  float* lf = (float*)d_ws;                                  // B*S*64 f32
  float* gf = lf + (size_t)BATCH * S_LEN * FDIM;             // B*S*64 f32

  // 1) fill 268 MB output with -inf (non-temporal b128 streaming stores)
  long n4 = (long)out_size / 4;
  fill_neginf_kernel<<<4096, 256, 0, stream>>>((v4f*)d_out, n4);

  // 2) feature projections (f32 WMMA)
  proj_kernel<<<(BATCH * S_LEN) / 16, 128, 0, stream>>>(x, Wl, Wg, lf, gf);

  // 3) score + stable top-k + scatter (f32 WMMA)
  hier_mask_kernel<<<dim3(S_LEN / 16, BATCH), 256, 0, stream>>>(lf, gf, out);
}